// AdaLoRA_58076547776863
// MI455X (gfx1250) — compile-verified
//
#include <hip/hip_runtime.h>

typedef __attribute__((ext_vector_type(2))) float v2f;
typedef __attribute__((ext_vector_type(8))) float v8f;

#define DIM   1024
#define RANK  16
#define NSLOT 2048          // B*K = 8*256
#define SCALE 0.25f         // 1.0 / sqrt(RANK)

// One 256-thread block (8 waves) per slot.
__global__ __launch_bounds__(256)
void AdaLoRA_gfx1250_kernel(const float* __restrict__ slots,
                            const int*   __restrict__ indices,
                            const float* __restrict__ downp,   // [4096][1024][16]
                            const float* __restrict__ upp,     // [4096][16][1024]
                            float*       __restrict__ out)     // [2048][1024]
{
    const int slot = blockIdx.x;          // b*256 + k
    const int tid  = threadIdx.x;
    const int wave = tid >> 5;            // 0..7
    const int lane = tid & 31;
    const int half = lane >> 4;           // 0 = lanes 0-15, 1 = lanes 16-31
    const int l16  = lane & 15;

    const int entry = indices[slot];
    const float* __restrict__ x = slots + (size_t)slot * DIM;
    const float* __restrict__ P = downp + (size_t)entry * DIM * RANK;  // P[d][r]
    const float* __restrict__ U = upp   + (size_t)entry * RANK * DIM;  // U[r][d]
    float* __restrict__ y = out + (size_t)slot * DIM;

    __shared__ float part[8][16];
    __shared__ float down[16];

    // ---------------- Phase 1: down[r] = sum_d x[d] * P[d][r] ----------------
    // WMMA f32 16x16x4:  C[r][n] += A[r][k]*B[k][n]
    //   A[r][k] = P[d0+k][r]   (transposed 4x16 chunk of P)
    //   B[k][n] = x[d0+k]      (broadcast across all 16 columns)
    // => every column of C holds the same partial down[] vector.
    v8f acc = {};
    #pragma unroll 4
    for (int i = 0; i < 32; ++i) {
        const int d0 = 4 * (wave + 8 * i);      // this wave's 4-wide d-chunk
        const int dA = d0 + 2 * half;           // lanes 0-15 -> K=0,1 ; 16-31 -> K=2,3
        v2f a;
        a.x = P[(size_t)dA       * RANK + l16]; // A[l16][2*half+0]
        a.y = P[(size_t)(dA + 1) * RANK + l16]; // A[l16][2*half+1]
        v2f b;
        b.x = x[dA];                            // B[2*half+0][*]
        b.y = x[dA + 1];                        // B[2*half+1][*]
        acc = __builtin_amdgcn_wmma_f32_16x16x4_f32(false, a, false, b,
                                                    (short)0, acc, false, false);
    }
    // C layout: VGPR j -> M=j (lanes 0-15), M=j+8 (lanes 16-31); all N equal.
    if (l16 == 0) {
        #pragma unroll
        for (int j = 0; j < 8; ++j) part[wave][half * 8 + j] = acc[j];
    }
    __syncthreads();
    if (tid < 16) {
        float s = 0.0f;
        #pragma unroll
        for (int w = 0; w < 8; ++w) s += part[w][tid];
        down[tid] = s;
    }
    __syncthreads();

    // ---------------- Phase 2: out[n] = sum_r down[r] * U[r][n] ----------------
    // WMMA f32 16x16x4:  C[m][n] += A[m][k]*B[k][n]
    //   A[m][k] = down[4s+k]          (broadcast across all 16 rows)
    //   B[k][n] = U[4s+k][tile*16+n]
    // => every row of C holds the same 16-wide output tile.
    v2f aup[4];
    #pragma unroll
    for (int s = 0; s < 4; ++s) {
        aup[s].x = down[4 * s + 2 * half];
        aup[s].y = down[4 * s + 2 * half + 1];
    }
    #pragma unroll 2
    for (int j = 0; j < 8; ++j) {
        const int tile = wave + 8 * j;          // 64 tiles of 16 outputs
        const int col  = tile * 16 + l16;
        v8f c = {};
        #pragma unroll
        for (int s = 0; s < 4; ++s) {
            const int r = 4 * s + 2 * half;     // lanes 0-15 -> rows 4s,4s+1 ; 16-31 -> 4s+2,4s+3
            v2f b;
            b.x = U[(size_t)r       * DIM + col];
            b.y = U[(size_t)(r + 1) * DIM + col];
            c = __builtin_amdgcn_wmma_f32_16x16x4_f32(false, aup[s], false, b,
                                                      (short)0, c, false, false);
        }
        // c[0], lanes 0-15: M=0, N=l16  -> out[tile*16 + l16]
        if (half == 0) y[col] = c[0] * SCALE;
    }
}

extern "C" void kernel_launch(void* const* d_in, const int* in_sizes, int n_in,
                              void* d_out, int out_size, void* d_ws, size_t ws_size,
                              hipStream_t stream) {
    const float* slots   = (const float*)d_in[0];
    const int*   indices = (const int*)  d_in[1];
    const float* downp   = (const float*)d_in[2];
    const float* upp     = (const float*)d_in[3];
    float*       out     = (float*)d_out;

    AdaLoRA_gfx1250_kernel<<<NSLOT, 256, 0, stream>>>(slots, indices, downp, upp, out);
}